// MemN2NDialog_26044681683703
// MI455X (gfx1250) — compile-verified
//
#include <hip/hip_runtime.h>
#include <math.h>

typedef __attribute__((ext_vector_type(2))) float v2f;
typedef __attribute__((ext_vector_type(8))) float v8f;

#define B_    16
#define M_    200
#define S_    32
#define C_    2048
#define V_    32000
#define D_    128
#define HOPS_ 3

// ---------------------------------------------------------------------------
// Kernel 1: story sentence embeddings  m[b,m,d] = sum_s embed_A[stories[b,m,s], d]
// one block per (b,m) row, 128 threads = one d each
// ---------------------------------------------------------------------------
__global__ void k_story_embed(const int* __restrict__ stories,
                              const float* __restrict__ embed_A,
                              float* __restrict__ m_ws) {
  int row = blockIdx.x;            // 0 .. B*M-1
  int d   = threadIdx.x;           // 0 .. 127
  const int* st = stories + row * S_;
  float acc = 0.f;
#pragma unroll
  for (int s = 0; s < S_; ++s) acc += embed_A[st[s] * D_ + d];
  m_ws[row * D_ + d] = acc;
}

// ---------------------------------------------------------------------------
// Kernel 2: query embedding + 3 memory hops. One block per batch b, 256 thr.
//   u0[d]   = sum_s embed_A[query[b,s], d]
//   hop:    attn = softmax(m . u);  o = attn . m;  u = u@H_w^T + H_b + o
// ---------------------------------------------------------------------------
__global__ void k_hops(const int* __restrict__ query,
                       const float* __restrict__ embed_A,
                       const float* __restrict__ H_w,
                       const float* __restrict__ H_b,
                       const float* __restrict__ m_ws,
                       float* __restrict__ u_ws) {
  __shared__ float u_s[D_];
  __shared__ float un_s[D_];
  __shared__ float attn_s[256];
  __shared__ float red[256];
  int b   = blockIdx.x;
  int tid = threadIdx.x;

  if (tid < D_) {
    const int* q = query + b * S_;
    float acc = 0.f;
#pragma unroll
    for (int s = 0; s < S_; ++s) acc += embed_A[q[s] * D_ + tid];
    u_s[tid] = acc;
  }
  __syncthreads();

  const float* mb = m_ws + b * M_ * D_;

  for (int hop = 0; hop < HOPS_; ++hop) {
    // --- scores: one memory row per thread ---
    float sc = -INFINITY;
    if (tid < M_) {
      const float* mr = mb + tid * D_;
      float acc = 0.f;
      for (int d = 0; d < D_; ++d) acc += mr[d] * u_s[d];
      sc = acc;
    }
    // --- softmax: max-reduce ---
    red[tid] = sc;
    __syncthreads();
    for (int off = 128; off > 0; off >>= 1) {
      if (tid < off) red[tid] = fmaxf(red[tid], red[tid + off]);
      __syncthreads();
    }
    float mx = red[0];
    __syncthreads();
    // --- exp + sum-reduce ---
    float e = (tid < M_) ? __expf(sc - mx) : 0.f;
    red[tid] = e;
    __syncthreads();
    for (int off = 128; off > 0; off >>= 1) {
      if (tid < off) red[tid] += red[tid + off];
      __syncthreads();
    }
    float inv = 1.f / red[0];
    __syncthreads();
    attn_s[tid] = e * inv;
    __syncthreads();
    // --- o = attn . m ;  u_new = H_w @ u + H_b + o ---
    if (tid < D_) {
      float od = 0.f;
      for (int mi = 0; mi < M_; ++mi) od += attn_s[mi] * mb[mi * D_ + tid];
      const float* hr = H_w + tid * D_;       // row d of H_w -> u@H_w^T
      float lin = H_b[tid];
      for (int k = 0; k < D_; ++k) lin += hr[k] * u_s[k];
      un_s[tid] = lin + od;
    }
    __syncthreads();
    if (tid < D_) u_s[tid] = un_s[tid];
    __syncthreads();
  }
  if (tid < D_) u_ws[b * D_ + tid] = u_s[tid];
}

// ---------------------------------------------------------------------------
// Kernel 3: uW[b, v] = sum_k u[b,k] * embed_W[v,k]    (16 x 32000, K=128)
// WMMA f32 16x16x4: each wave owns one 16(b) x 16(v) tile, K-loop of 32 steps.
// Fragment layouts per CDNA5 ISA 7.12.2:
//   A 16x4 : lane<16 -> {K=k0,k0+1}, lane>=16 -> {K=k0+2,k0+3}, M = lane&15
//   B 4x16 : same K split, N = lane&15   (B = embed_W^T, so B[k][n]=embed_W[v0+n][k])
//   C/D    : VGPR g -> M = g + 8*(lane>>4), N = lane&15
// ---------------------------------------------------------------------------
__global__ void k_uw_wmma(const float* __restrict__ u_ws,
                          const float* __restrict__ embed_W,
                          float* __restrict__ uW_ws) {
  int wave = threadIdx.x >> 5;
  int lane = threadIdx.x & 31;
  int tile = blockIdx.x * 8 + wave;        // 0 .. 1999
  int v0   = tile * 16;
  int r    = lane & 15;
  int hi   = lane >> 4;

  const float* Ap = u_ws    + r * D_        + 2 * hi;  // u[r][k0 + 2*hi ..]
  const float* Bp = embed_W + (v0 + r) * D_ + 2 * hi;  // embed_W[v0+r][k0 + 2*hi ..]

  v8f acc = {0.f, 0.f, 0.f, 0.f, 0.f, 0.f, 0.f, 0.f};
#pragma unroll
  for (int k0 = 0; k0 < D_; k0 += 4) {
    v2f a  = *(const v2f*)(Ap + k0);
    v2f bf = *(const v2f*)(Bp + k0);
    acc = __builtin_amdgcn_wmma_f32_16x16x4_f32(false, a, false, bf,
                                                (short)0, acc, false, false);
  }
#pragma unroll
  for (int g = 0; g < 8; ++g)
    uW_ws[(g + 8 * hi) * V_ + v0 + r] = acc[g];
}

// ---------------------------------------------------------------------------
// Kernel 4: logits[b,c] = sum_s uW[b,candidates[c,s]] + sum_s uW[b,E[b,c,s]]
// uW[b,:] (128 KB) staged through LDS in two 64 KB chunks -> ds_load gathers.
// grid = (C/256, B), 256 threads, one candidate per thread.
// ---------------------------------------------------------------------------
__global__ void k_logits(const int* __restrict__ E,
                         const int* __restrict__ candidates,
                         const float* __restrict__ uW_ws,
                         float* __restrict__ out) {
  constexpr int VCH = 16000;               // 64000 B static LDS chunk
  __shared__ float uWs[VCH];
  int b = blockIdx.y;
  int c = blockIdx.x * blockDim.x + threadIdx.x;
  const float* uWb = uW_ws + b * V_;
  const int*   cc  = candidates + c * S_;
  const int*   ec  = E + (b * C_ + c) * S_;

  float acc = 0.f;
  for (int p = 0; p < V_; p += VCH) {
    __syncthreads();
    for (int i = threadIdx.x; i < VCH; i += blockDim.x) uWs[i] = uWb[p + i];
    __syncthreads();
#pragma unroll
    for (int s = 0; s < S_; ++s) {
      int t = cc[s] - p;
      if (t >= 0 && t < VCH) acc += uWs[t];
    }
#pragma unroll
    for (int s = 0; s < S_; ++s) {
      int t = ec[s] - p;
      if (t >= 0 && t < VCH) acc += uWs[t];
    }
  }
  out[b * C_ + c] = acc;
}

// ---------------------------------------------------------------------------
extern "C" void kernel_launch(void* const* d_in, const int* in_sizes, int n_in,
                              void* d_out, int out_size, void* d_ws, size_t ws_size,
                              hipStream_t stream) {
  const int*   stories    = (const int*)d_in[0];    // [B,M,S]
  const int*   query      = (const int*)d_in[1];    // [B,S]
  const int*   E          = (const int*)d_in[2];    // [B,C,S]
  const int*   candidates = (const int*)d_in[3];    // [C,S]
  const float* embed_A    = (const float*)d_in[4];  // [V,D]
  const float* embed_W    = (const float*)d_in[5];  // [V,D]
  const float* H_w        = (const float*)d_in[6];  // [D,D]
  const float* H_b        = (const float*)d_in[7];  // [D]
  float* out = (float*)d_out;                       // [B,C]

  float* ws    = (float*)d_ws;
  float* m_ws  = ws;                    // B*M*D   = 409600 f
  float* u_ws  = m_ws + B_ * M_ * D_;   // B*D     =   2048 f
  float* uW_ws = u_ws + B_ * D_;        // B*V     = 512000 f  (~3.7 MB total)

  k_story_embed<<<B_ * M_, D_, 0, stream>>>(stories, embed_A, m_ws);
  k_hops<<<B_, 256, 0, stream>>>(query, embed_A, H_w, H_b, m_ws, u_ws);
  k_uw_wmma<<<(V_ / 16) / 8, 256, 0, stream>>>(u_ws, embed_W, uW_ws);
  dim3 g4(C_ / 256, B_);
  k_logits<<<g4, 256, 0, stream>>>(E, candidates, uW_ws, out);
}